// GaANNet_pooling_38397007626857
// MI455X (gfx1250) — compile-verified
//
#include <hip/hip_runtime.h>
#include <hip/hip_bf16.h>
#include <math.h>

// ---------------------------------------------------------------------------
// GaAN net for MI455X (gfx1250, wave32). Dense GEMMs use the native fp32
// matrix instruction V_WMMA_F32_16X16X4_F32 via an LDS-tiled GEMM
// (32x64 block tile, BK=32, global_load_b128 staging -> ds_load fragments).
// The workload is memory/atomic bound (~5GB edge traffic vs 23.3 TB/s HBM),
// so fp32 WMMA keeps reference precision at zero throughput cost.
// Segment softmax over edges uses f32 global atomics.
// ---------------------------------------------------------------------------

typedef float v2f __attribute__((ext_vector_type(2)));
typedef float v8f __attribute__((ext_vector_type(8)));

static constexpr int NN      = 6000;    // nodes
static constexpr int NE      = 96000;   // edges
static constexpr int CF      = 256;     // C_FEAT
static constexpr int SS      = 128;     // S (H=W)
static constexpr int DLOI    = 128;     // DIM_LOI
static constexpr int FPOOL   = 2048;    // DIM_LOI * N_PTS0/2
static constexpr int DHFC    = 1024;    // DIM_HOUR_FC
static constexpr int DHOUT   = 128;     // DIM_HOUR_OUT
static constexpr int EHID    = 32;      // EDGE_HID
static constexpr int HID     = 256;     // HIDDEN
static constexpr int NHEADS  = 4;

__device__ __forceinline__ int imin(int a, int b) { return a < b ? a : b; }

// ---------------------------------------------------------------------------
// LDS-tiled fp32 WMMA GEMM:  C(MxN) = act(A(MxK) * B(KxN) + bias)
// ATRANS: A stored K-major (K x M), i.e. A[k*lda + m] (the NCHW 1x1 conv).
// Block: 256 threads = 8 waves; tile 32(M) x 64(N); K-step 32.
// OOB M-rows / N-cols are only address-clamped: garbage values feed C rows/
// cols that the guarded store never writes. Only the K tail is zero-masked.
// Fragment layouts per CDNA5 ISA 7.12.2:
//   A 16x4 : lanes 0-15 -> (K0,K1) of row M=lane ; lanes 16-31 -> (K2,K3)
//   B 4x16 : vgpr0 = rows K0/K2, vgpr1 = rows K1/K3, N = lane&15
//   C 16x16: vgpr r -> rows r (lanes 0-15) and r+8 (lanes 16-31)
// ---------------------------------------------------------------------------
template <bool ATRANS, int ACT>   // ACT: 0 = none, 1 = ReLU
__global__ __launch_bounds__(256)
void gemm_f32_wmma(const float* __restrict__ A, const float* __restrict__ B,
                   const float* __restrict__ bias, float* __restrict__ C,
                   int M, int N, int K, int lda, int ldb, int ldc) {
  __shared__ float As[32][36];   // [k][m], row pad keeps 16B alignment
  __shared__ float Bs[32][68];   // [k][n]

  const int t    = threadIdx.x;
  const int lane = t & 31;
  const int wave = t >> 5;                       // 0..7
  const int m0b  = blockIdx.y * 32;
  const int n0b  = blockIdx.x * 64;
  const int mw   = (wave >> 2) * 16;             // wave tile inside block
  const int nw   = (wave & 3) * 16;

  const int l  = lane & 15;
  const int hi = lane >> 4;                      // 0 or 1 (half-wave)

  v8f acc0 = {}, acc1 = {};

  for (int k0 = 0; k0 < K; k0 += 32) {
    const bool full_k = (k0 + 32 <= K);

    // ---- stage A tile: As[k][m] ----
    if (ATRANS) {
      const int kk = t >> 3;                     // 0..31
      const int mq = (t & 7) * 4;                // 0..28
      const int gk = k0 + kk;
      if (full_k && (m0b + 32 <= M)) {
        const float4 v = *(const float4*)&A[(size_t)gk * lda + (m0b + mq)];
        *(float4*)&As[kk][mq] = v;
        if (k0 + 32 < K)
          __builtin_prefetch(&A[(size_t)(gk + 32) * lda + (m0b + mq)], 0, 3);
      } else {
#pragma unroll
        for (int i = 0; i < 4; ++i) {
          const int gm = imin(m0b + mq + i, M - 1);
          const float v = A[(size_t)imin(gk, K - 1) * lda + gm];
          As[kk][mq + i] = (gk < K) ? v : 0.f;
        }
      }
    } else {
      const int m  = t >> 3;                     // 0..31
      const int kq = (t & 7) * 4;                // 0..28
      const int gm = imin(m0b + m, M - 1);       // clamp: garbage rows unstored
      if (full_k) {
        const float4 v = *(const float4*)&A[(size_t)gm * lda + (k0 + kq)];
        As[kq + 0][m] = v.x; As[kq + 1][m] = v.y;
        As[kq + 2][m] = v.z; As[kq + 3][m] = v.w;
        if (k0 + 32 < K)
          __builtin_prefetch(&A[(size_t)gm * lda + (k0 + 32 + kq)], 0, 3);
      } else {
#pragma unroll
        for (int i = 0; i < 4; ++i) {
          const int gk = k0 + kq + i;
          const float v = A[(size_t)gm * lda + imin(gk, K - 1)];
          As[kq + i][m] = (gk < K) ? v : 0.f;
        }
      }
    }

    // ---- stage B tile: Bs[k][n] ----
    {
      const int kq = t >> 4;                     // 0..15 (two k halves)
      const int nq = (t & 15) * 4;               // 0..60
      const bool full_n = (n0b + 64 <= N);
#pragma unroll
      for (int half = 0; half < 32; half += 16) {
        const int kk = kq + half;
        const int gk = k0 + kk;
        if (full_k && full_n) {
          const float4 v = *(const float4*)&B[(size_t)gk * ldb + (n0b + nq)];
          *(float4*)&Bs[kk][nq] = v;
        } else {
#pragma unroll
          for (int i = 0; i < 4; ++i) {
            const int gn = imin(n0b + nq + i, N - 1);
            const float v = B[(size_t)imin(gk, K - 1) * ldb + gn];
            Bs[kk][nq + i] = (gk < K) ? v : 0.f;
          }
        }
      }
    }

    __syncthreads();

    // ---- 8 WMMAs per K-tile, two independent accumulator chains ----
#pragma unroll
    for (int kk = 0; kk < 32; kk += 8) {
      const int ka0 = kk + hi * 2;
      v2f a0; a0.x = As[ka0][mw + l];     a0.y = As[ka0 + 1][mw + l];
      v2f b0; b0.x = Bs[ka0][nw + l];     b0.y = Bs[ka0 + 1][nw + l];
      const int ka1 = kk + 4 + hi * 2;
      v2f a1; a1.x = As[ka1][mw + l];     a1.y = As[ka1 + 1][mw + l];
      v2f b1; b1.x = Bs[ka1][nw + l];     b1.y = Bs[ka1 + 1][nw + l];
      acc0 = __builtin_amdgcn_wmma_f32_16x16x4_f32(false, a0, false, b0,
                                                   (short)0, acc0, false, false);
      acc1 = __builtin_amdgcn_wmma_f32_16x16x4_f32(false, a1, false, b1,
                                                   (short)0, acc1, false, false);
    }

    __syncthreads();
  }

  const v8f acc = acc0 + acc1;
#pragma unroll
  for (int r = 0; r < 8; ++r) {
    const int row = m0b + mw + hi * 8 + r;
    const int col = n0b + nw + l;
    if (row < M && col < N) {
      float v = acc[r];
      if (bias) v += bias[col];
      if (ACT == 1) v = fmaxf(v, 0.f);
      C[(size_t)row * ldc + col] = v;
    }
  }
}

// ---------------------------------------------------------------------------
// Line pooling: per-node bilinear sampling of loiT (HW x C, channel-minor so
// the 128 channel-threads coalesce), max over 3 normal offsets, pairwise max
// along the line -> feat[n][c*16+s]
// ---------------------------------------------------------------------------
__global__ __launch_bounds__(128)
void pooling_kernel(const float* __restrict__ loiT, const float* __restrict__ x,
                    float* __restrict__ feat) {
  const int n = blockIdx.x;
  const int c = threadIdx.x;           // channel 0..127
  const float ux = (x[n * 16 + 7]  + 1.f) * 0.5f * (float)SS;
  const float uy = (x[n * 16 + 8]  + 1.f) * 0.5f * (float)SS;
  const float vx = (x[n * 16 + 9]  + 1.f) * 0.5f * (float)SS;
  const float vy = (x[n * 16 + 10] + 1.f) * 0.5f * (float)SS;
  const float dx = vx - ux, dy = vy - uy;
  const float inv = 1.f / fmaxf(sqrtf(dx * dx + dy * dy), 1e-12f);
  const float nx =  dy * inv;          // line_norm = (dir_y, -dir_x)
  const float ny = -dx * inv;

  for (int i = 0; i < 32; i += 2) {
    float pairmax = -INFINITY;
#pragma unroll
    for (int ii = 0; ii < 2; ++ii) {
      const float t0 = (float)(i + ii) * (1.f / 31.f);
      const float cx = ux * t0 + vx * (1.f - t0);
      const float cy = uy * t0 + vy * (1.f - t0);
#pragma unroll
      for (int j = -1; j <= 1; ++j) {
        const float px = cx + nx * (float)j;   // t1 in {-1,0,1}
        const float py = cy + ny * (float)j;
        const float px0 = fminf(fmaxf(floorf(px), 0.f), 127.f);
        const float py0 = fminf(fmaxf(floorf(py), 0.f), 127.f);
        const float px1 = fminf(px0 + 1.f, 127.f);
        const float py1 = fminf(py0 + 1.f, 127.f);
        const int ix0 = (int)px0, iy0 = (int)py0;
        const int ix1 = (int)px1, iy1 = (int)py1;
        const float s =
            loiT[((size_t)iy0 * SS + ix0) * DLOI + c] * ((py1 - py) * (px1 - px)) +
            loiT[((size_t)iy1 * SS + ix0) * DLOI + c] * ((py - py0) * (px1 - px)) +
            loiT[((size_t)iy0 * SS + ix1) * DLOI + c] * ((py1 - py) * (px - px0)) +
            loiT[((size_t)iy1 * SS + ix1) * DLOI + c] * ((py - py0) * (px - px0));
        pairmax = fmaxf(pairmax, s);
      }
    }
    feat[(size_t)n * FPOOL + c * 16 + (i >> 1)] = pairmax;
  }
}

// h_node = [x @ Wnode + b , pooled]
__global__ void hnode_kernel(const float* __restrict__ x,
                             const float* __restrict__ W, const float* __restrict__ b,
                             const float* __restrict__ pooled, float* __restrict__ h0) {
  const long i = (long)blockIdx.x * blockDim.x + threadIdx.x;
  if (i >= (long)NN * 256) return;
  const int n = (int)(i / 256), c = (int)(i % 256);
  if (c < DHOUT) {
    float acc = b[c];
#pragma unroll
    for (int k = 0; k < 16; ++k) acc += x[n * 16 + k] * W[k * DHOUT + c];
    h0[i] = acc;
  } else {
    h0[i] = pooled[(size_t)n * DHOUT + (c - DHOUT)];
  }
}

// h_edge = edge_attr @ Wedge + b
__global__ void hedge_kernel(const float* __restrict__ ea,
                             const float* __restrict__ W, const float* __restrict__ b,
                             float* __restrict__ he) {
  const long i = (long)blockIdx.x * blockDim.x + threadIdx.x;
  if (i >= (long)NE * EHID) return;
  const int e = (int)(i / EHID), c = (int)(i % EHID);
  float acc = b[c];
#pragma unroll
  for (int k = 0; k < 8; ++k) acc += ea[e * 8 + k] * W[k * EHID + c];
  he[i] = acc;
}

__global__ void fill_kernel(float* p, float v, long n) {
  const long i = (long)blockIdx.x * blockDim.x + threadIdx.x;
  if (i < n) p[i] = v;
}
__global__ void fill2d_kernel(float* p, int ld, int rows, int cols, float v) {
  const long i = (long)blockIdx.x * blockDim.x + threadIdx.x;
  if (i >= (long)rows * cols) return;
  p[(size_t)(i / cols) * ld + (i % cols)] = v;
}
__global__ void copy2d_kernel(float* dst, int ldd, const float* src, int lds,
                              int rows, int cols) {
  const long i = (long)blockIdx.x * blockDim.x + threadIdx.x;
  if (i >= (long)rows * cols) return;
  const int r = (int)(i / cols), c = (int)(i % cols);
  dst[(size_t)r * ldd + c] = src[(size_t)r * lds + c];
}

// float atomic max via sign-aware int ordering (-> global_atomic_{max_i32,min_u32})
__device__ __forceinline__ void atomicMaxFloat(float* addr, float v) {
  if (v >= 0.f) atomicMax((int*)addr, __float_as_int(v));
  else          atomicMin((unsigned int*)addr, __float_as_uint(v));
}

// a = leaky(ai[dst] + aj[src] + e @ WaE + ba); segment-max into amax
__global__ void edge_attn_a(const float* __restrict__ ai, const float* __restrict__ aj,
                            const float* __restrict__ he, const float* __restrict__ WaE,
                            const float* __restrict__ ba,
                            const int* __restrict__ src, const int* __restrict__ dst,
                            float* __restrict__ abuf, float* __restrict__ amax, int dout) {
  const long i = (long)blockIdx.x * blockDim.x + threadIdx.x;
  if (i >= (long)NE * dout) return;
  const int e = (int)(i / dout), c = (int)(i % dout);
  const int s = src[e], d = dst[e];
  float a = ba[c] + ai[(size_t)d * dout + c] + aj[(size_t)s * dout + c];
#pragma unroll
  for (int k = 0; k < EHID; ++k) a += he[e * EHID + k] * WaE[k * dout + c];
  a = (a >= 0.f) ? a : 0.2f * a;
  abuf[i] = a;
  atomicMaxFloat(&amax[(size_t)d * dout + c], a);
}

// w = exp(a - amax[dst]); segment-sum into denom
__global__ void edge_attn_b(const int* __restrict__ dst, const float* __restrict__ amax,
                            float* __restrict__ abuf, float* __restrict__ denom, int dout) {
  const long i = (long)blockIdx.x * blockDim.x + threadIdx.x;
  if (i >= (long)NE * dout) return;
  const int e = (int)(i / dout), c = (int)(i % dout);
  const float w = __expf(abuf[i] - amax[(size_t)dst[e] * dout + c]);
  abuf[i] = w;
  atomicAdd(&denom[(size_t)dst[e] * dout + c], w);
}

// out[dst] += (w/denom[dst]) * xfc[src]
__global__ void edge_attn_c(const int* __restrict__ src, const int* __restrict__ dst,
                            const float* __restrict__ abuf, const float* __restrict__ denom,
                            const float* __restrict__ xfc, float* __restrict__ zhead,
                            int ldz, int dout) {
  const long i = (long)blockIdx.x * blockDim.x + threadIdx.x;
  if (i >= (long)NE * dout) return;
  const int e = (int)(i / dout), c = (int)(i % dout);
  const int s = src[e], d = dst[e];
  const float w = abuf[i] / (denom[(size_t)d * dout + c] + 1e-16f);
  atomicAdd(&zhead[(size_t)d * ldz + c], w * xfc[(size_t)s * dout + c]);
}

__global__ void sigmoid_kernel(const float* __restrict__ in, float* __restrict__ out, long n) {
  const long i = (long)blockIdx.x * blockDim.x + threadIdx.x;
  if (i < n) out[i] = 1.f / (1.f + __expf(-in[i]));
}

// ---------------------------------------------------------------------------
static inline void gemm(hipStream_t st, bool atrans, int act,
                        const float* A, const float* B, const float* bias, float* C,
                        int M, int N, int K, int lda, int ldb, int ldc) {
  dim3 g((N + 63) / 64, (M + 31) / 32), b(256);
  if (atrans) {
    if (act) gemm_f32_wmma<true, 1><<<g, b, 0, st>>>(A, B, bias, C, M, N, K, lda, ldb, ldc);
    else     gemm_f32_wmma<true, 0><<<g, b, 0, st>>>(A, B, bias, C, M, N, K, lda, ldb, ldc);
  } else {
    if (act) gemm_f32_wmma<false, 1><<<g, b, 0, st>>>(A, B, bias, C, M, N, K, lda, ldb, ldc);
    else     gemm_f32_wmma<false, 0><<<g, b, 0, st>>>(A, B, bias, C, M, N, K, lda, ldb, ldc);
  }
}
static inline dim3 g1(long n) { return dim3((unsigned)((n + 255) / 256)); }

extern "C" void kernel_launch(void* const* d_in, const int* in_sizes, int n_in,
                              void* d_out, int out_size, void* d_ws, size_t ws_size,
                              hipStream_t stream) {
  // Input ordering: top-level dict insertion order {features,x,edge_index,
  // edge_attr,params}; params tree-flattened jax-style (dict keys sorted,
  // lists in order, each linear layer as (b,w)).
  const float* features = (const float*)d_in[0];  // (256,128,128) K-major
  const float* x        = (const float*)d_in[1];  // (6000,16)
  const int*   eidx     = (const int*)  d_in[2];  // (2,96000)
  const float* eattr    = (const float*)d_in[3];  // (96000,8)
  auto P = [&](int i) { return (const float*)d_in[4 + i]; };
  // params layout (offsets relative to index 4):
  //  conv_in : fc.b=0 fc.w=1 heads@2   (head h: +8h -> Wa.b,Wa.w,Wn.b,Wn.w,Wpi.b,Wpi.w,Wpj.b,Wpj.w)
  //  conv_out: fc.b=34 fc.w=35 heads@36
  //  fc_edge : b=68 w=69 ; fc_hour: 70..75 ; fc_loi: b=76 w=77 ; fc_node: b=78 w=79
  //  hiddens[0]: fc.b=80 fc.w=81 heads@82 ; hiddens[1]: fc.b=114 fc.w=115 heads@116
  const int* src = eidx;
  const int* dst = eidx + NE;

  // ---- workspace layout (floats) ----
  float* ws = (float*)d_ws;
  size_t off = 0;
  auto take = [&](size_t n) { float* p = ws + off; off += n; return p; };
  float* hedge  = take((size_t)NE * EHID);
  float* pooled = take((size_t)NN * DHOUT);
  float* h0     = take((size_t)NN * HID);
  float* hA     = take((size_t)NN * HID);
  float* hB     = take((size_t)NN * HID);
  float* xfc    = take((size_t)NN * HID);
  float* pni    = take((size_t)NN * HID);
  float* pnj    = take((size_t)NN * HID);
  float* ani    = take((size_t)NN * HID);
  float* anj    = take((size_t)NN * HID);
  float* amax   = take((size_t)NN * HID);
  float* denom  = take((size_t)NN * HID);
  float* z      = take((size_t)NN * 1280);
  float* outpre = take((size_t)NN * 2);
  // big region: phase-1 buffers aliased over the per-head edge buffer
  float* big    = take(26673152);               // max(abuf, loiT+feat+hh1+hh2)
  float* loiT   = big;                          // (16384,128)
  float* featP  = loiT + (size_t)SS * SS * DLOI;
  float* hh1    = featP + (size_t)NN * FPOOL;
  float* hh2    = hh1 + (size_t)NN * DHFC;
  float* abuf   = big;                          // (96000,256) per head

  // ---- phase 1: loi conv (A transposed), pooling, hour MLP ----
  gemm(stream, true, 0, features, P(77), P(76), loiT, SS * SS, DLOI, CF, SS * SS, DLOI, DLOI);
  pooling_kernel<<<NN, 128, 0, stream>>>(loiT, x, featP);
  gemm(stream, false, 1, featP, P(71), P(70), hh1, NN, DHFC, FPOOL, FPOOL, DHFC, DHFC);
  gemm(stream, false, 1, hh1,   P(73), P(72), hh2, NN, DHFC, DHFC, DHFC, DHFC, DHFC);
  gemm(stream, false, 0, hh2,   P(75), P(74), pooled, NN, DHOUT, DHFC, DHFC, DHOUT, DHOUT);
  hnode_kernel<<<g1((long)NN * 256), 256, 0, stream>>>(x, P(79), P(78), pooled, h0);
  hedge_kernel<<<g1((long)NE * EHID), 256, 0, stream>>>(eattr, P(69), P(68), hedge);

  // ---- GaAN layers ----
  struct LayerIdx { int fcb, fcw, hb, din, dout; const float* hin; float* hout; };
  LayerIdx layers[4] = {
      {0,   1,   2,   HID, HID, h0, hA},   // conv_in  (din = 2*DIM_HOUR_OUT = 256)
      {80,  81,  82,  HID, HID, hA, hB},   // hiddens[0]
      {114, 115, 116, HID, HID, hB, hA},   // hiddens[1]
      {34,  35,  36,  HID, 2,   hA, outpre}// conv_out
  };

  for (int L = 0; L < 4; ++L) {
    const LayerIdx& ly = layers[L];
    const int dout = ly.dout, din = ly.din;
    const int ldz = din + NHEADS * dout;
    const long nd = (long)NN * dout;
    const long ed = (long)NE * dout;

    copy2d_kernel<<<g1((long)NN * din), 256, 0, stream>>>(z, ldz, ly.hin, din, NN, din);

    for (int h = 0; h < NHEADS; ++h) {
      const int hb = ly.hb + 8 * h;
      const float *Wab = P(hb), *Waw = P(hb + 1);
      const float *Wnb = P(hb + 2), *Wnw = P(hb + 3);
      const float *Wpib = P(hb + 4), *Wpiw = P(hb + 5);
      const float *Wpjb = P(hb + 6), *Wpjw = P(hb + 7);
      // per-node projections (gather commutes with linear maps)
      gemm(stream, false, 0, ly.hin, Wnw,  Wnb,  xfc, NN, dout, din,  din,  dout, dout);
      gemm(stream, false, 0, xfc,    Wpiw, Wpib, pni, NN, dout, dout, dout, dout, dout);
      gemm(stream, false, 0, xfc,    Wpjw, Wpjb, pnj, NN, dout, dout, dout, dout, dout);
      gemm(stream, false, 0, pni, Waw,                       nullptr, ani, NN, dout, dout, dout, dout, dout);
      gemm(stream, false, 0, pnj, Waw + (size_t)dout * dout, nullptr, anj, NN, dout, dout, dout, dout, dout);
      // edge softmax (segment max / sum / weighted scatter via f32 atomics)
      fill_kernel<<<g1(nd), 256, 0, stream>>>(amax, -INFINITY, nd);
      fill_kernel<<<g1(nd), 256, 0, stream>>>(denom, 0.f, nd);
      fill2d_kernel<<<g1(nd), 256, 0, stream>>>(z + din + h * dout, ldz, NN, dout, 0.f);
      const float* WaE = Waw + (size_t)2 * dout * dout;
      edge_attn_a<<<g1(ed), 256, 0, stream>>>(ani, anj, hedge, WaE, Wab, src, dst, abuf, amax, dout);
      edge_attn_b<<<g1(ed), 256, 0, stream>>>(dst, amax, abuf, denom, dout);
      edge_attn_c<<<g1(ed), 256, 0, stream>>>(src, dst, abuf, denom, xfc,
                                              z + din + h * dout, ldz, dout);
    }
    gemm(stream, false, 0, z, P(ly.fcw), P(ly.fcb), ly.hout, NN, dout, ldz, ldz, dout, dout);
  }

  sigmoid_kernel<<<g1((long)NN * 2), 256, 0, stream>>>(outpre, (float*)d_out, (long)NN * 2);
}